// GATConv_13091060319065
// MI455X (gfx1250) — compile-verified
//
#include <hip/hip_runtime.h>
#include <math.h>

typedef float v2f __attribute__((ext_vector_type(2)));
typedef float v8f __attribute__((ext_vector_type(8)));
typedef unsigned int u32x4 __attribute__((ext_vector_type(4)));
typedef int i32x4 __attribute__((ext_vector_type(4)));
typedef int i32x8 __attribute__((ext_vector_type(8)));

#define FIN 128
#define HD  128      // H*D
#define H   4
#define D   32
#define C   16
#define HC  64       // H*C
#define NEG_SLOPE 0.2f
#define LDS_ROW 132  // 128 + 4 pad DWORDs (bank-conflict-free fragment reads)

// ---------------------------------------------------------------------------
// 0) init: rst = bias (broadcast), ypred = 0, m = -0.0f, s = 0
// ---------------------------------------------------------------------------
__global__ void gat_init_kernel(float* __restrict__ rst, float* __restrict__ ypred,
                                float* __restrict__ m, float* __restrict__ s,
                                const float* __restrict__ bias, int N) {
    long long i = (long long)blockIdx.x * blockDim.x + threadIdx.x;
    long long total = (long long)N * HD;
    if (i >= total) return;
    int j = (int)(i & (HD - 1));        // column 0..127
    long long n = i >> 7;               // node
    rst[i] = bias[j];
    if (j < HC)  ypred[n * HC + j] = 0.0f;
    if (j < H) {
        m[n * H + j] = -0.0f;           // 0x80000000: identity for the float max trick
        s[n * H + j] = 0.0f;
    }
}

// ---------------------------------------------------------------------------
// 1) ft = feat @ fc_w   via V_WMMA_F32_16X16X4_F32 (wave32, fp32 end-to-end)
//    A tile (16x128 f32) staged into LDS by the Tensor Data Mover with
//    descriptor-driven row padding (+4 DW -> stride 132) for conflict-free
//    ds_load_b64 fragment reads. One wave -> one 16x16 output tile.
// ---------------------------------------------------------------------------
__global__ __launch_bounds__(256)
void gat_gemm_f32_wmma_kernel(const float* __restrict__ feat,
                              const float* __restrict__ fcw,
                              float* __restrict__ ft, int N) {
    __shared__ float lds_a[16 * LDS_ROW];

    const int wave = threadIdx.x >> 5;          // 0..7 -> output col tile
    const int lane = threadIdx.x & 31;
    const int row0 = blockIdx.x * 16;

    // --- wave 0: TDM load of the A tile into LDS ------------------------
    if (threadIdx.x < 32) {
        unsigned long long ga = (unsigned long long)(const void*)(feat + (size_t)row0 * FIN);
        unsigned ldsOff = (unsigned)(unsigned long long)(void*)lds_a;
        int rows = N - row0; if (rows > 16) rows = 16;   // OOB rows zero-filled

        // D# group 0: count=1, lds_addr, 57-bit global_addr, type=2 (image)
        u32x4 g0;
        g0[0] = 1u;
        g0[1] = ldsOff;
        g0[2] = (unsigned)(ga & 0xFFFFFFFFull);
        g0[3] = (unsigned)((ga >> 32) & 0x1FFFFFFull) | (2u << 30);

        // D# group 1: data_size=2 (4B), pad_enable, pad_interval=6 (128 DW),
        // pad_amount=3 (4 DW), tensor_dim0=128, tensor_dim1=rows,
        // tile_dim0=128, tile_dim1=16, tensor_dim0_stride=128
        unsigned long long q0 = (2ull << 16) | (1ull << 20) | (6ull << 22) | (3ull << 25)
                              | ((unsigned long long)(FIN & 0xFFFF) << 48);
        unsigned long long q1 = ((unsigned long long)FIN >> 16)
                              | ((unsigned long long)(unsigned)rows << 16)
                              | ((unsigned long long)FIN << 48);
        unsigned long long q2 = 16ull
                              | ((unsigned long long)FIN << 32);
        unsigned long long q3 = 0ull;
        i32x8 g1;
        g1[0] = (int)(unsigned)q0; g1[1] = (int)(q0 >> 32);
        g1[2] = (int)(unsigned)q1; g1[3] = (int)(q1 >> 32);
        g1[4] = (int)(unsigned)q2; g1[5] = (int)(q2 >> 32);
        g1[6] = (int)(unsigned)q3; g1[7] = (int)(q3 >> 32);
        i32x4 gz4 = {0, 0, 0, 0};                // groups 2/3 unused (2-D tile)
        i32x8 gz8 = {0, 0, 0, 0, 0, 0, 0, 0};    // extra operand, zeroed

        __builtin_amdgcn_tensor_load_to_lds(g0, g1, gz4, gz4, gz8, 0);
        __builtin_amdgcn_s_wait_tensorcnt(0);
    }
    __syncthreads();

    // --- WMMA main loop --------------------------------------------------
    const int n0    = wave * 16;
    const int am    = lane & 15;                 // A fragment M index
    const int khalf = (lane >> 4) * 2;           // A/B fragment K base (0 or 2)
    const int bn    = lane & 15;                 // B fragment N index

    const float* ap    = &lds_a[am * LDS_ROW + khalf];
    const float* bcolp = fcw + (size_t)khalf * HD + n0 + bn;

    v8f acc = {0.f, 0.f, 0.f, 0.f, 0.f, 0.f, 0.f, 0.f};

    #pragma unroll 4
    for (int k0 = 0; k0 < FIN; k0 += 4) {
        v2f a = *(const v2f*)(ap + k0);          // ds_load_b64, conflict-free
        v2f b;
        b[0] = bcolp[(size_t)k0 * HD];
        b[1] = bcolp[(size_t)(k0 + 1) * HD];
        acc = __builtin_amdgcn_wmma_f32_16x16x4_f32(
                  false, a, false, b, (short)0, acc, false, false);
    }

    // C/D layout: VGPR v, lanes 0-15 -> row v, lanes 16-31 -> row v+8; col = lane&15
    const int rbase = row0 + ((lane >> 4) * 8);
    const int cc    = n0 + (lane & 15);
    if (row0 + 16 <= N) {                        // fast path: full tile
        float* out = ft + (size_t)rbase * HD + cc;
        #pragma unroll
        for (int v = 0; v < 8; ++v) out[(size_t)v * HD] = acc[v];
    } else {                                     // tail tile
        #pragma unroll
        for (int v = 0; v < 8; ++v) {
            int r = rbase + v;
            if (r < N) ft[(size_t)r * HD + cc] = acc[v];
        }
    }
}

// ---------------------------------------------------------------------------
// 2) el[n,h] = sum_d ft[n,h,d]*attn_l[h,d] ; er likewise
// ---------------------------------------------------------------------------
__global__ void gat_elr_kernel(const float* __restrict__ ft,
                               const float* __restrict__ attn_l,
                               const float* __restrict__ attn_r,
                               float* __restrict__ el, float* __restrict__ er, int N) {
    int i = blockIdx.x * blockDim.x + threadIdx.x;     // over N*H
    if (i >= N * H) return;
    int h = i & (H - 1);
    long long n = i >> 2;
    const float* f  = ft + n * HD + h * D;
    const float* al = attn_l + h * D;
    const float* ar = attn_r + h * D;
    float sl = 0.f, sr = 0.f;
    #pragma unroll 8
    for (int d = 0; d < D; ++d) {
        float v = f[d];
        sl = fmaf(v, al[d], sl);
        sr = fmaf(v, ar[d], sr);
    }
    el[i] = sl;
    er[i] = sr;
}

// ---------------------------------------------------------------------------
// atomic float max via int/uint ordering trick (init must be -0.0f)
// ---------------------------------------------------------------------------
__device__ __forceinline__ void atomicMaxF32(float* addr, float v) {
    if (v >= 0.0f) atomicMax((int*)addr, __float_as_int(v));
    else           atomicMin((unsigned int*)addr, __float_as_uint(v));
}

__device__ __forceinline__ float leaky(float x) {
    return x > 0.0f ? x : NEG_SLOPE * x;
}

// ---------------------------------------------------------------------------
// 3) per (edge,h): m[dst,h] = max(m[dst,h], leaky(el[src,h]+er[dst,h]))
// ---------------------------------------------------------------------------
__global__ void gat_edge_max_kernel(const float* __restrict__ el,
                                    const float* __restrict__ er,
                                    const int* __restrict__ src,
                                    const int* __restrict__ dst,
                                    float* __restrict__ m, int E) {
    long long i = (long long)blockIdx.x * blockDim.x + threadIdx.x;
    if (i >= (long long)E * H) return;
    int h = (int)(i & (H - 1));
    int e = (int)(i >> 2);
    int sN = src[e], dN = dst[e];
    float val = leaky(el[sN * H + h] + er[dN * H + h]);
    atomicMaxF32(&m[dN * H + h], val);
}

// ---------------------------------------------------------------------------
// 4) per (edge,h): s[dst,h] += exp(e - m[dst,h])
// ---------------------------------------------------------------------------
__global__ void gat_edge_sum_kernel(const float* __restrict__ el,
                                    const float* __restrict__ er,
                                    const int* __restrict__ src,
                                    const int* __restrict__ dst,
                                    const float* __restrict__ m,
                                    float* __restrict__ s, int E) {
    long long i = (long long)blockIdx.x * blockDim.x + threadIdx.x;
    if (i >= (long long)E * H) return;
    int h = (int)(i & (H - 1));
    int e = (int)(i >> 2);
    int sN = src[e], dN = dst[e];
    float val = leaky(el[sN * H + h] + er[dN * H + h]);
    float ee = __expf(val - m[dN * H + h]);
    atomicAdd(&s[dN * H + h], ee);
}

// ---------------------------------------------------------------------------
// 5) wave-per-edge scatter:
//    rst[dst,h,d]  += ft[src,h,d] * a[h]     (128 f32 per edge, 4 lane-sweeps)
//    ypred[dst,h,c] += y[src,c]   * a[h]     (64 f32 per edge, 2 lane-sweeps)
// ---------------------------------------------------------------------------
__global__ __launch_bounds__(256)
void gat_edge_scatter_kernel(const float* __restrict__ ft,
                             const float* __restrict__ y,
                             const float* __restrict__ el,
                             const float* __restrict__ er,
                             const float* __restrict__ m,
                             const float* __restrict__ s,
                             const int* __restrict__ src,
                             const int* __restrict__ dst,
                             float* __restrict__ rst,
                             float* __restrict__ ypred, int E) {
    long long gtid = (long long)blockIdx.x * blockDim.x + threadIdx.x;
    int edge = (int)(gtid >> 5);
    int lane = threadIdx.x & 31;
    if (edge >= E) return;

    int sN = src[edge], dN = dst[edge];

    float a[H];
    #pragma unroll
    for (int h = 0; h < H; ++h) {
        float e  = leaky(el[sN * H + h] + er[dN * H + h]);
        float ee = __expf(e - m[dN * H + h]);
        a[h] = ee / fmaxf(s[dN * H + h], 1e-9f);
    }

    const float* fsrc = ft + (size_t)sN * HD;
    float*       rdst = rst + (size_t)dN * HD;
    #pragma unroll
    for (int it = 0; it < HD / 32; ++it) {
        int j = it * 32 + lane;                 // j = h*32 + d
        atomicAdd(&rdst[j], fsrc[j] * a[j >> 5]);
    }

    const float* ysrc = y + (size_t)sN * C;
    float*       ydst = ypred + (size_t)dN * HC;
    #pragma unroll
    for (int it = 0; it < HC / 32; ++it) {
        int j = it * 32 + lane;                 // j = h*16 + c
        atomicAdd(&ydst[j], ysrc[j & (C - 1)] * a[j >> 4]);
    }
}

// ---------------------------------------------------------------------------
// launcher
// inputs: 0 feat(N,128) 1 y(N,16) 2 fc_w(128,128) 3 attn_l(4,32) 4 attn_r(4,32)
//         5 bias(128) 6 src(E) 7 dst(E)
// out: rst (N*128) then y_pred (N*64), both f32
// ---------------------------------------------------------------------------
extern "C" void kernel_launch(void* const* d_in, const int* in_sizes, int n_in,
                              void* d_out, int out_size, void* d_ws, size_t ws_size,
                              hipStream_t stream) {
    const float* feat   = (const float*)d_in[0];
    const float* y      = (const float*)d_in[1];
    const float* fcw    = (const float*)d_in[2];
    const float* attn_l = (const float*)d_in[3];
    const float* attn_r = (const float*)d_in[4];
    const float* bias   = (const float*)d_in[5];
    const int*   src    = (const int*)d_in[6];
    const int*   dst    = (const int*)d_in[7];

    const int N = in_sizes[0] / FIN;
    const int E = in_sizes[6];

    float* rst   = (float*)d_out;                       // N*128
    float* ypred = (float*)d_out + (size_t)N * HD;      // N*64

    float* ws = (float*)d_ws;
    float* ft = ws;                                     // N*128
    float* el = ft + (size_t)N * HD;                    // N*4
    float* er = el + (size_t)N * H;                     // N*4
    float* m  = er + (size_t)N * H;                     // N*4
    float* s  = m  + (size_t)N * H;                     // N*4

    // 0) init outputs + softmax state
    {
        long long total = (long long)N * HD;
        int blocks = (int)((total + 255) / 256);
        gat_init_kernel<<<blocks, 256, 0, stream>>>(rst, ypred, m, s, bias, N);
    }
    // 1) projection GEMM (TDM -> LDS -> WMMA f32)
    {
        int blocks = (N + 15) / 16;
        gat_gemm_f32_wmma_kernel<<<blocks, 256, 0, stream>>>(feat, fcw, ft, N);
    }
    // 2) el / er
    {
        int blocks = (N * H + 255) / 256;
        gat_elr_kernel<<<blocks, 256, 0, stream>>>(ft, attn_l, attn_r, el, er, N);
    }
    // 3) segment max
    {
        long long total = (long long)E * H;
        int blocks = (int)((total + 255) / 256);
        gat_edge_max_kernel<<<blocks, 256, 0, stream>>>(el, er, src, dst, m, E);
    }
    // 4) segment sum of exp
    {
        long long total = (long long)E * H;
        int blocks = (int)((total + 255) / 256);
        gat_edge_sum_kernel<<<blocks, 256, 0, stream>>>(el, er, src, dst, m, s, E);
    }
    // 5) weighted scatter (wave per edge)
    {
        long long total = (long long)E * 32;
        int blocks = (int)((total + 255) / 256);
        gat_edge_scatter_kernel<<<blocks, 256, 0, stream>>>(
            ft, y, el, er, m, s, src, dst, rst, ypred, E);
    }
}